// AttentionWithContext_42820823941246
// MI455X (gfx1250) — compile-verified
//
#include <hip/hip_runtime.h>
#include <math.h>

typedef __attribute__((ext_vector_type(2))) float v2f;
typedef __attribute__((ext_vector_type(8))) float v8f;

#define NN 2048
#define DD 256
#define BB 8

// ---------------------------------------------------------------------------
// Kernel 1: h[b,n,d] = sum_k x[b,k,n] * W[d,k]        (B,N,D) output
// WMMA f32 16x16x4. A = xt tile (16n x 256k) held in 128 VGPRs, W streamed.
// grid = B*16 blocks, 256 threads (8 waves); wave handles one 16-row n-tile.
// ---------------------------------------------------------------------------
__global__ __launch_bounds__(256) void k_h(const float* __restrict__ x,
                                           const float* __restrict__ W,
                                           float* __restrict__ h) {
    const int lane = threadIdx.x & 31;
    const int wave = threadIdx.x >> 5;
    const int half = lane >> 4;
    const int l15  = lane & 15;
    const int b    = blockIdx.x >> 4;
    const int nBase = (blockIdx.x & 15) * 128 + wave * 16;

    const float* xB = x + (size_t)b * DD * NN;

    // A[m=n][k] = x[b,k,nBase+m]; lane holds K pair {2*half, 2*half+1} per slice
    v2f A[64];
#pragma unroll
    for (int ks = 0; ks < 64; ++ks) {
        int k0 = ks * 4 + 2 * half;
        A[ks].x = xB[(size_t)k0 * NN + nBase + l15];
        A[ks].y = xB[(size_t)(k0 + 1) * NN + nBase + l15];
    }

    for (int dt = 0; dt < 16; ++dt) {
        // B[k][d] = W[(dt*16 + d)*256 + k] : per-lane float2, contiguous in k
        const float* wPtr = W + (dt * 16 + l15) * DD + 2 * half;
        v8f c = {};
#pragma unroll
        for (int ks = 0; ks < 64; ++ks) {
            v2f bv = *(const v2f*)(wPtr + ks * 4);
            c = __builtin_amdgcn_wmma_f32_16x16x4_f32(
                false, A[ks], false, bv, (short)0, c, false, false);
        }
#pragma unroll
        for (int r = 0; r < 8; ++r) {
            int n = nBase + r + 8 * half;
            h[((size_t)b * NN + n) * DD + dt * 16 + l15] = c[r];
        }
    }
}

// ---------------------------------------------------------------------------
// Kernel 2: per (b,n): s1 = h.a1, sj = h.a2 (init), t = h.a3
// one wave per row; grid = (B*N)/8 blocks of 256 threads.
// ---------------------------------------------------------------------------
__global__ __launch_bounds__(256) void k_vec(const float* __restrict__ h,
                                             const float* __restrict__ a,
                                             float* __restrict__ tbuf,
                                             float* __restrict__ s1buf,
                                             float* __restrict__ sjbuf) {
    const int lane = threadIdx.x & 31;
    const int row  = blockIdx.x * 8 + (threadIdx.x >> 5);  // 0..B*N-1
    const float* r = h + (size_t)row * DD;
    float p1 = 0.f, p2 = 0.f, p3 = 0.f;
#pragma unroll
    for (int kk = 0; kk < 8; ++kk) {
        int k = lane + 32 * kk;
        float v = r[k];
        p1 += v * a[k];
        p2 += v * a[DD + k];
        p3 += v * a[2 * DD + k];
    }
#pragma unroll
    for (int off = 1; off < 32; off <<= 1) {
        p1 += __shfl_xor(p1, off, 32);
        p2 += __shfl_xor(p2, off, 32);
        p3 += __shfl_xor(p3, off, 32);
    }
    if (lane == 0) {
        s1buf[row] = p1;
        sjbuf[row] = p2;   // initial sj = h.a2 ; Ca3 added by k_ca3
        tbuf[row]  = p3;
    }
}

// ---------------------------------------------------------------------------
// Kernel 3: Ca3[b,i] = sum_j (adj[i,j]!=0 && i!=j) * (h_i . h_j) * t[b,j]
// sj[b,i] += Ca3[b,i].
// grid = B*128 blocks (one per (b, i-tile of 16)); 8 waves split the j range.
// sim tiles via chained v_wmma_f32_16x16x4_f32, never materialized.
// ---------------------------------------------------------------------------
__global__ __launch_bounds__(256) void k_ca3(const float* __restrict__ h,
                                             const int* __restrict__ adj,
                                             const float* __restrict__ tbuf,
                                             float* __restrict__ sjbuf) {
    const int lane = threadIdx.x & 31;
    const int wave = threadIdx.x >> 5;
    const int half = lane >> 4;
    const int l15  = lane & 15;
    const int b     = blockIdx.x >> 7;
    const int iBase = (blockIdx.x & 127) * 16;

    const float* hB = h + (size_t)b * NN * DD;
    const float* tB = tbuf + (size_t)b * NN;

    // A = H_I tile (16 x 256), kept in registers across the whole j loop
    v2f A[64];
    {
        const float* aPtr = hB + (iBase + l15) * DD + 2 * half;
#pragma unroll
        for (int ks = 0; ks < 64; ++ks) A[ks] = *(const v2f*)(aPtr + ks * 4);
    }

    float ca[8];
#pragma unroll
    for (int r = 0; r < 8; ++r) ca[r] = 0.f;

    for (int q = 0; q < 16; ++q) {
        const int jBase = (wave + 8 * q) * 16;   // 8 waves interleave 128 tiles
        // B = H_J^T : B[k][n] = h[jBase+n][k] -> same float2 pattern
        const float* bPtr = hB + (jBase + l15) * DD + 2 * half;
        v8f c = {};
#pragma unroll
        for (int ks = 0; ks < 64; ++ks) {
            v2f bv = *(const v2f*)(bPtr + ks * 4);
            c = __builtin_amdgcn_wmma_f32_16x16x4_f32(
                false, A[ks], false, bv, (short)0, c, false, false);
        }
        const int j = jBase + l15;
        const float tj = tB[j];
#pragma unroll
        for (int r = 0; r < 8; ++r) {
            int i = iBase + r + 8 * half;
            bool ok = (adj[(size_t)i * NN + j] != 0) && (i != j);
            ca[r] += ok ? c[r] * tj : 0.f;
        }
    }

    // reduce over the 16 lanes (j) of each half-wave
#pragma unroll
    for (int r = 0; r < 8; ++r) {
#pragma unroll
        for (int off = 1; off < 16; off <<= 1)
            ca[r] += __shfl_xor(ca[r], off, 32);
    }

    __shared__ float red[8][16];
    if (l15 == 0) {
#pragma unroll
        for (int r = 0; r < 8; ++r) red[wave][r + 8 * half] = ca[r];
    }
    __syncthreads();
    if (threadIdx.x < 16) {
        float s = 0.f;
#pragma unroll
        for (int w = 0; w < 8; ++w) s += red[w][threadIdx.x];
        sjbuf[(size_t)b * NN + iBase + threadIdx.x] += s;
    }
}

// ---------------------------------------------------------------------------
// Kernel 4: scores row softmax.  grid = B*N blocks, 256 threads, 8 j's each.
// ---------------------------------------------------------------------------
__global__ __launch_bounds__(256) void k_soft(const int* __restrict__ adj,
                                              const float* __restrict__ s1buf,
                                              const float* __restrict__ sjbuf,
                                              float* __restrict__ out) {
    const int lane = threadIdx.x & 31;
    const int wave = threadIdx.x >> 5;
    const int b = blockIdx.x >> 11;
    const int i = blockIdx.x & (NN - 1);

    const float si = s1buf[(size_t)b * NN + i];
    const float* sjB = sjbuf + (size_t)b * NN;
    const int* adjR = adj + (size_t)i * NN;

    float e[8];
    float mx = -3.4e38f;
#pragma unroll
    for (int k = 0; k < 8; ++k) {
        int j = threadIdx.x + 256 * k;
        float v;
        if (adjR[j] != 0) {
            float z = si + sjB[j];
            v = z > 0.f ? z : 0.2f * z;   // leaky_relu(0.2)
        } else {
            v = -9.0e15f;
        }
        e[k] = v;
        mx = fmaxf(mx, v);
    }

    __shared__ float smax[8];
    __shared__ float ssum[8];
#pragma unroll
    for (int off = 1; off < 32; off <<= 1) mx = fmaxf(mx, __shfl_xor(mx, off, 32));
    if (lane == 0) smax[wave] = mx;
    __syncthreads();
    float MX = smax[0];
#pragma unroll
    for (int w = 1; w < 8; ++w) MX = fmaxf(MX, smax[w]);

    float s = 0.f;
#pragma unroll
    for (int k = 0; k < 8; ++k) {
        e[k] = expf(e[k] - MX);
        s += e[k];
    }
#pragma unroll
    for (int off = 1; off < 32; off <<= 1) s += __shfl_xor(s, off, 32);
    if (lane == 0) ssum[wave] = s;
    __syncthreads();
    float S = 0.f;
#pragma unroll
    for (int w = 0; w < 8; ++w) S += ssum[w];
    const float inv = 1.0f / S;

    float* o = out + ((size_t)b * NN + i) * NN;
#pragma unroll
    for (int k = 0; k < 8; ++k) o[threadIdx.x + 256 * k] = e[k] * inv;
}

// ---------------------------------------------------------------------------
extern "C" void kernel_launch(void* const* d_in, const int* in_sizes, int n_in,
                              void* d_out, int out_size, void* d_ws, size_t ws_size,
                              hipStream_t stream) {
    const float* x   = (const float*)d_in[0];  // (B, DIN, N)
    const int*   adj = (const int*)d_in[1];    // (N, N)
    const float* W   = (const float*)d_in[2];  // (DIN, DIN)
    const float* a   = (const float*)d_in[3];  // (3*DIN,)
    float* out = (float*)d_out;                // (B, N, N)

    float* ws = (float*)d_ws;
    float* h    = ws;                               // B*N*D  = 4,194,304 floats
    float* tbuf = h + (size_t)BB * NN * DD;         // B*N
    float* s1   = tbuf + (size_t)BB * NN;           // B*N
    float* sj   = s1 + (size_t)BB * NN;             // B*N

    k_h   <<<dim3(BB * 16),  dim3(256), 0, stream>>>(x, W, h);
    k_vec <<<dim3(BB * NN / 8), dim3(256), 0, stream>>>(h, a, tbuf, s1, sj);
    k_ca3 <<<dim3(BB * 128), dim3(256), 0, stream>>>(h, adj, tbuf, sj);
    k_soft<<<dim3(BB * NN),  dim3(256), 0, stream>>>(adj, s1, sj, out);
}